// Q_Fun_5815385719436
// MI455X (gfx1250) — compile-verified
//
#include <hip/hip_runtime.h>
#include <hip/hip_bf16.h>

typedef __attribute__((ext_vector_type(2))) float v2f;
typedef __attribute__((ext_vector_type(8))) float v8f;

#define NNODES  50000
#define NEDGES  800000
#define HDIM    64
#define TSTEPS  4
#define LDSTR   66              // padded LDS row stride (floats), breaks bank conflicts
#define NTILES  (NNODES / 16)   // 3125, exact

// ---------------------------------------------------------------- utilities
__global__ void zero_f(float* __restrict__ p, int n) {
    int i = blockIdx.x * blockDim.x + threadIdx.x;
    if (i < n) p[i] = 0.0f;
}

// deg[n] = #edges with idx==n ; s[n] = sum of edge_w over those edges
__global__ void edge_hist(const int* __restrict__ idx, const float* __restrict__ ew,
                          float* __restrict__ deg, float* __restrict__ s) {
    int e = blockIdx.x * blockDim.x + threadIdx.x;
    if (e < NEDGES) {
        int n = idx[e];
        atomicAdd(&deg[n], 1.0f);
        atomicAdd(&s[n], ew[e]);
    }
}

// v1[t][h] = W1[t][h][0] ; v3[t][h] = sum_i W3[t][h][i] * relu(W4[t][i][0])
__global__ void layer_consts(const float* __restrict__ W1, const float* __restrict__ W3,
                             const float* __restrict__ W4,
                             float* __restrict__ v1, float* __restrict__ v3) {
    int i = threadIdx.x;            // 0..255 == t*64+h
    int t = i >> 6, h = i & 63;
    v1[i] = W1[i];
    const float* w3row = W3 + (size_t)(t * 64 + h) * 64;
    const float* w4    = W4 + t * 64;
    float acc = 0.0f;
    for (int k = 0; k < 64; ++k) acc += w3row[k] * fmaxf(w4[k], 0.0f);
    v3[i] = acc;
}

// ---------------------------------------------------------------- WMMA GEMM
// acc[tt] (16x16 f32 tiles, tt = output-col tile) += tileLDS(16x64) * B(64x64)^T
// B is row-major [64 out][64 in]; computes D[m][o] = sum_k A[m][k]*B[o][k].
__device__ __forceinline__ void tile_gemm64(const float* __restrict__ tp,
                                            const float* __restrict__ B,
                                            int lane, v8f acc[4]) {
    const int m    = lane & 15;
    const int half = lane >> 4;
    for (int kk = 0; kk < 16; ++kk) {          // K = 64 in steps of 4
        const int kcol = kk * 4 + half * 2;
        v2f a = *(const v2f*)(tp + m * LDSTR + kcol);           // A[m][kcol..+1]
        for (int tt = 0; tt < 4; ++tt) {
            const int o = tt * 16 + m;                          // output column
            v2f b = *(const v2f*)(B + (size_t)o * HDIM + kcol); // B[o][kcol..+1]
            acc[tt] = __builtin_amdgcn_wmma_f32_16x16x4_f32(
                false, a, false, b, (short)0, acc[tt], false, false);
        }
    }
}

// ------------------------------------------------ fused T-step S2V + readout
__global__ __launch_bounds__(256) void s2v_fused(
    const float* __restrict__ mu,  const float* __restrict__ x,
    const float* __restrict__ W2,  const float* __restrict__ W7,
    const float* __restrict__ W5,
    const float* __restrict__ deg, const float* __restrict__ s,
    const float* __restrict__ v1,  const float* __restrict__ v3,
    float* __restrict__ muFin, float* __restrict__ outp)
{
    __shared__ float lds[8][16 * LDSTR];
    const int wave = threadIdx.x >> 5;
    const int lane = threadIdx.x & 31;
    const int tile = blockIdx.x * 8 + wave;
    if (tile >= NTILES) return;                 // wave-uniform exit
    float* tp = lds[wave];
    const int rowBase = tile * 16;
    const int m = lane & 15, half = lane >> 4;

    // load 16x64 mu tile into LDS (coalesced float4 reads)
    for (int i = 0; i < 8; ++i) {
        int q = i * 32 + lane;                  // float4 index in tile
        int r = q >> 4, col = (q & 15) << 2;
        float4 g = *(const float4*)(mu + (size_t)(rowBase + r) * HDIM + col);
        float* dst = tp + r * LDSTR + col;
        dst[0] = g.x; dst[1] = g.y; dst[2] = g.z; dst[3] = g.w;
    }

    // T fused message-passing steps, tile stays in LDS
    for (int t = 0; t < TSTEPS; ++t) {
        v8f acc[4] = {};
        tile_gemm64(tp, W2 + (size_t)t * HDIM * HDIM, lane, acc);
        for (int v = 0; v < 8; ++v) {
            int mr  = half * 8 + v;
            int row = rowBase + mr;
            float dv = deg[row], xv = x[row], sv = s[row];
            for (int tt = 0; tt < 4; ++tt) {
                int col = tt * 16 + m;
                float val = dv * acc[tt][v]
                          + xv * v1[t * HDIM + col]
                          + sv * v3[t * HDIM + col];
                tp[mr * LDSTR + col] = fmaxf(val, 0.0f);
            }
        }
    }

    // store final mu tile (for the graph-pool column sums)
    for (int i = 0; i < 8; ++i) {
        int q = i * 32 + lane;
        int r = q >> 4, col = (q & 15) << 2;
        const float* sp = tp + r * LDSTR + col;
        float4 g = make_float4(sp[0], sp[1], sp[2], sp[3]);
        *(float4*)(muFin + (size_t)(rowBase + r) * HDIM + col) = g;
    }

    // nodes_vec term: outp[n] = sum_o relu((W7 mu_fin[n])[o]) * W5[64+o]
    {
        v8f acc[4] = {};
        tile_gemm64(tp, W7, lane, acc);
        float part[8];
        for (int v = 0; v < 8; ++v) {
            float p = 0.0f;
            for (int tt = 0; tt < 4; ++tt) {
                int col = tt * 16 + m;
                p += fmaxf(acc[tt][v], 0.0f) * W5[64 + col];
            }
            part[v] = p;
        }
        // reduce across the 16 lanes of each half-wave (xor stays in-group)
        for (int off = 1; off < 16; off <<= 1)
            for (int v = 0; v < 8; ++v)
                part[v] += __shfl_xor(part[v], off, 32);
        if (m == 0)
            for (int v = 0; v < 8; ++v)
                outp[rowBase + half * 8 + v] = part[v];
    }
}

// ---------------------------------------------------------------- pooling
__global__ __launch_bounds__(256) void colsum(const float* __restrict__ muFin,
                                              float* __restrict__ P) {
    __shared__ float sp[256];
    int tid = threadIdx.x;
    int col = tid & 63, rg = tid >> 6;
    float acc = 0.0f;
    for (int r = blockIdx.x * 4 + rg; r < NNODES; r += gridDim.x * 4)
        acc += muFin[(size_t)r * HDIM + col];
    sp[tid] = acc;
    __syncthreads();
    if (tid < 64)
        atomicAdd(&P[col], sp[tid] + sp[tid + 64] + sp[tid + 128] + sp[tid + 192]);
}

__global__ void pool_dot(const float* __restrict__ P, const float* __restrict__ W5,
                         float* __restrict__ c) {
    __shared__ float sp[64];
    int tid = threadIdx.x;              // 64 threads
    sp[tid] = fmaxf(P[tid], 0.0f) * W5[tid];
    __syncthreads();
    for (int off = 32; off > 0; off >>= 1) {
        if (tid < off) sp[tid] += sp[tid + off];
        __syncthreads();
    }
    if (tid == 0) c[0] = sp[0];
}

__global__ void finalize(const float* __restrict__ outp, const float* __restrict__ c,
                         float* __restrict__ out) {
    int n = blockIdx.x * blockDim.x + threadIdx.x;
    if (n < NNODES) out[n] = c[0] + outp[n];
}

// ---------------------------------------------------------------- launcher
extern "C" void kernel_launch(void* const* d_in, const int* in_sizes, int n_in,
                              void* d_out, int out_size, void* d_ws, size_t ws_size,
                              hipStream_t stream) {
    (void)n_in; (void)out_size; (void)ws_size;
    const float* mu  = (const float*)d_in[0];
    const float* x   = (const float*)d_in[1];
    const int*   ei  = (const int*)d_in[2];     // [2,E]; row 1 = idx
    const float* ew  = (const float*)d_in[3];
    const float* W1  = (const float*)d_in[4];
    const float* W2  = (const float*)d_in[5];
    const float* W3  = (const float*)d_in[6];
    const float* W4  = (const float*)d_in[7];
    const float* W5  = (const float*)d_in[8];
    const float* W7  = (const float*)d_in[9];
    float* out = (float*)d_out;

    const int E = in_sizes[2] / 2;              // = NEDGES
    const int* idx = ei + E;                    // edge_index[1]

    // workspace layout (floats)
    float* ws    = (float*)d_ws;
    float* muFin = ws;                          // 3,200,000
    float* deg   = ws + 3200000;                // 50,000
    float* sArr  = ws + 3250000;                // 50,000
    float* P     = ws + 3300000;                // 64   (contiguous with deg/s)
    float* v1    = ws + 3300064;                // 256
    float* v3    = ws + 3300320;                // 256
    float* cS    = ws + 3300576;                // 1
    float* outp  = ws + 3300608;                // 50,000

    // zero the accumulation region deg..s..P (contiguous 100064 floats)
    zero_f<<<(100064 + 255) / 256, 256, 0, stream>>>(deg, 100064);
    edge_hist<<<(E + 255) / 256, 256, 0, stream>>>(idx, ew, deg, sArr);
    layer_consts<<<1, 256, 0, stream>>>(W1, W3, W4, v1, v3);
    s2v_fused<<<(NTILES + 7) / 8, 256, 0, stream>>>(mu, x, W2, W7, W5,
                                                    deg, sArr, v1, v3, muFin, outp);
    colsum<<<128, 256, 0, stream>>>(muFin, P);
    pool_dot<<<1, 64, 0, stream>>>(P, W5, cS);
    finalize<<<(NNODES + 255) / 256, 256, 0, stream>>>(outp, cS, out);
}